// Graphsage_Model_27539330302492
// MI455X (gfx1250) — compile-verified
//
#include <hip/hip_runtime.h>
#include <hip/hip_bf16.h>
#include <cstdint>

// ---------------------------------------------------------------------------
// GCN + query-MLP + candidate scoring for MI455X (gfx1250, wave32).
// Heavy GEMMs use V_WMMA_F32_16X16X4_F32 (full fp32 precision, matches the
// fp32 reference). A-tiles are staged with the gfx1250 async global->LDS DMA
// path (GLOBAL_LOAD_ASYNC_TO_LDS_B128, ASYNCcnt) -- no VGPR round-trip.
// Edge aggregation uses global_atomic_add_f32 resolved in the 192MB L2
// (dest arrays are 25.6MB/12.8MB). Self-loop contribution is fused into the
// epilogue, removing 50k*128 atomics.
// ---------------------------------------------------------------------------

typedef __attribute__((ext_vector_type(2))) float v2f;
typedef __attribute__((ext_vector_type(8))) float v8f;
typedef __attribute__((ext_vector_type(4))) int   v4i;

#if defined(__HIP_DEVICE_COMPILE__) && __has_builtin(__builtin_amdgcn_global_load_async_to_lds_b128) && __has_builtin(__builtin_amdgcn_s_wait_asynccnt)
#define USE_ASYNC_LDS 1
#else
#define USE_ASYNC_LDS 0
#endif

#if USE_ASYNC_LDS
// Flat->AS(1)/AS(3) pointer conversion. For LDS the flat address keeps the
// workgroup-relative LDS byte offset in addr[31:0] (ISA aperture mapping),
// so truncation recovers the DS address.
__device__ __forceinline__ __attribute__((address_space(1))) v4i*
to_global_b128(const void* p) {
  return (__attribute__((address_space(1))) v4i*)(uintptr_t)p;
}
__device__ __forceinline__ __attribute__((address_space(3))) v4i*
to_lds_b128(void* p) {
  return (__attribute__((address_space(3))) v4i*)(uint32_t)(uintptr_t)p;
}
#endif

// ---------------------------------------------------------------------------
// Generic fp32 WMMA GEMM:  Y[M x N_OUT] = X[M x K_DIM] * W[K_DIM x N_OUT]
// Workgroup: N_OUT/16 waves, covers 64 rows x N_OUT cols.
// Each wave owns a 16-wide column slab and 4 row-subtiles (16 rows each),
// so every B fragment is reused by 4 WMMAs. A is staged via async DMA into
// LDS in 64-wide K chunks, padded (+4) so rows stay 16B-aligned for B128
// writes AND column-strided fragment reads hit distinct banks.
// K_DIM must be a multiple of 64; N_OUT a multiple of 16.
// ---------------------------------------------------------------------------
template <int K_DIM, int N_OUT>
__global__ void gemm_f32_wmma(const float* __restrict__ X,
                              const float* __restrict__ W,
                              float* __restrict__ Y, int M) {
  constexpr int WAVES = N_OUT / 16;
  constexpr int NTHREADS = WAVES * 32;
  constexpr int ROWS = 64;   // rows per workgroup
  constexpr int KC = 64;     // K chunk staged in LDS
  constexpr int PAD = 4;     // keep rows 16B aligned; banks (4*rr+cc)%64

  __shared__ float As[ROWS][KC + PAD];

  const int tid = threadIdx.x;
  const int wave = tid >> 5;
  const int lane = tid & 31;
  const int half = lane >> 4;  // 0: lanes 0-15, 1: lanes 16-31
  const int r = lane & 15;
  const int rowBase = blockIdx.x * ROWS;
  const int nBase = wave * 16;

  v8f c0 = {}, c1 = {}, c2 = {}, c3 = {};

  for (int kc = 0; kc < K_DIM; kc += KC) {
    __syncthreads();
    // Stage A[rowBase..rowBase+63][kc..kc+63]: 1024 16B segments.
    // Trip count uniform across the block -> EXEC stays all-ones for WMMA.
#if USE_ASYNC_LDS
    for (int seg = tid; seg < ROWS * (KC / 4); seg += NTHREADS) {
      int rr = seg >> 4;          // 16 segments per row
      int c4 = (seg & 15) << 2;   // starting float column
      int grow = rowBase + rr;
      if (grow >= M) grow = M - 1;  // clamp tail (stores guarded later)
      const float* g = X + (long)grow * K_DIM + kc + c4;
      __builtin_amdgcn_global_load_async_to_lds_b128(
          to_global_b128((const void*)g), to_lds_b128((void*)&As[rr][c4]), 0, 0);
    }
    __builtin_amdgcn_s_wait_asynccnt(0);
#else
    for (int idx = tid; idx < ROWS * KC; idx += NTHREADS) {
      int rr = idx >> 6;
      int cc = idx & (KC - 1);
      int grow = rowBase + rr;
      if (grow >= M) grow = M - 1;
      As[rr][cc] = X[(long)grow * K_DIM + kc + cc];
    }
#endif
    __syncthreads();

    for (int k = 0; k < KC; k += 4) {
      // B fragment (4x16): lane half 0 holds rows k,k+1; half 1 rows k+2,k+3.
      const long kg = (long)(kc + k + 2 * half);
      v2f b;
      b.x = W[kg * N_OUT + nBase + r];
      b.y = W[(kg + 1) * N_OUT + nBase + r];
      // A fragments (16x4) for the 4 row-subtiles, from LDS.
      const int ak = k + 2 * half;
      {
        v2f a; a.x = As[r][ak];      a.y = As[r][ak + 1];
        c0 = __builtin_amdgcn_wmma_f32_16x16x4_f32(false, a, false, b, (short)0, c0, false, false);
      }
      {
        v2f a; a.x = As[16 + r][ak]; a.y = As[16 + r][ak + 1];
        c1 = __builtin_amdgcn_wmma_f32_16x16x4_f32(false, a, false, b, (short)0, c1, false, false);
      }
      {
        v2f a; a.x = As[32 + r][ak]; a.y = As[32 + r][ak + 1];
        c2 = __builtin_amdgcn_wmma_f32_16x16x4_f32(false, a, false, b, (short)0, c2, false, false);
      }
      {
        v2f a; a.x = As[48 + r][ak]; a.y = As[48 + r][ak + 1];
        c3 = __builtin_amdgcn_wmma_f32_16x16x4_f32(false, a, false, b, (short)0, c3, false, false);
      }
    }
  }

  // Store D: VGPR j holds row (8*half + j), column nBase + r.
  v8f acc[4] = {c0, c1, c2, c3};
#pragma unroll
  for (int m = 0; m < 4; ++m) {
    const int row0 = rowBase + m * 16 + half * 8;
#pragma unroll
    for (int j = 0; j < 8; ++j) {
      const int grow = row0 + j;
      if (grow < M) Y[(long)grow * N_OUT + nBase + r] = acc[m][j];
    }
  }
}

// ---------------------------------------------------------------------------
// Degree / normalization
// ---------------------------------------------------------------------------
__global__ void k_init_deg(float* deg, int n) {
  int i = blockIdx.x * blockDim.x + threadIdx.x;
  if (i < n) deg[i] = 1.0f;  // self-loop
}

__global__ void k_count_deg(const int* __restrict__ dst, int E, float* deg) {
  int e = blockIdx.x * blockDim.x + threadIdx.x;
  if (e < E) atomicAdd(&deg[dst[e]], 1.0f);  // exact: counts << 2^24
}

__global__ void k_make_dis(const float* __restrict__ deg, float* dis, int n) {
  int i = blockIdx.x * blockDim.x + threadIdx.x;
  if (i < n) dis[i] = rsqrtf(deg[i]);  // deg >= 1 always
}

// ---------------------------------------------------------------------------
// Edge scatter: AGG[dst] += dis[src]*dis[dst] * XW[src]
// Feature index fastest: one edge per 32-lane group (F is 64 or 128) ->
// uniform src/dst loads, coalesced atomics into one destination row.
// ---------------------------------------------------------------------------
template <int F>
__global__ void k_scatter(const int* __restrict__ src, const int* __restrict__ dst,
                          int E, const float* __restrict__ dis,
                          const float* __restrict__ XW, float* __restrict__ AGG) {
  long long idx = (long long)blockIdx.x * blockDim.x + threadIdx.x;
  long long total = (long long)E * F;
  if (idx >= total) return;
  int e = (int)(idx >> (F == 128 ? 7 : 6));
  int f = (int)(idx & (F - 1));
  int s = src[e];
  int d = dst[e];
  float nrm = dis[s] * dis[d];
  atomicAdd(&AGG[(long)d * F + f], nrm * XW[(long)s * F + f]);
}

// ---------------------------------------------------------------------------
// Epilogue: OUT = AGG + XW/deg (fused self-loop, dis[i]^2 = 1/deg[i]) + bias,
// optional sigmoid. In-place on AGG is safe (1:1 element map).
// ---------------------------------------------------------------------------
template <int F, bool SIG>
__global__ void k_finish_conv(const float* __restrict__ AGG, const float* __restrict__ XW,
                              const float* __restrict__ deg, const float* __restrict__ bias,
                              float* __restrict__ OUT, int n) {
  long long idx = (long long)blockIdx.x * blockDim.x + threadIdx.x;
  if (idx >= (long long)n * F) return;
  int i = (int)(idx >> (F == 128 ? 7 : 6));
  int f = (int)(idx & (F - 1));
  float v = AGG[idx] + XW[idx] / deg[i] + bias[f];
  if (SIG) v = 1.0f / (1.0f + __expf(-v));
  OUT[idx] = v;
}

// ---------------------------------------------------------------------------
// BatchNorm (inference) + sigmoid on z (B x 256); folds sn_b1 in.
// ---------------------------------------------------------------------------
__global__ void k_bn_sigmoid(float* z, const float* __restrict__ b,
                             const float* __restrict__ gamma, const float* __restrict__ beta,
                             const float* __restrict__ mean, const float* __restrict__ var,
                             int total) {
  int idx = blockIdx.x * blockDim.x + threadIdx.x;
  if (idx >= total) return;
  int f = idx & 255;
  float v = z[idx] + b[f];
  v = (v - mean[f]) * rsqrtf(var[f] + 1e-5f) * gamma[f] + beta[f];
  z[idx] = 1.0f / (1.0f + __expf(-v));
}

// ---------------------------------------------------------------------------
// score[b,k] = dot(qraw[b]+sn_b2, G[cand[b,k]]) over D=64 (sn_b2 folded here)
// ---------------------------------------------------------------------------
__global__ void k_score(const float* __restrict__ qraw, const float* __restrict__ b2,
                        const float* __restrict__ G, const int* __restrict__ cand,
                        float* __restrict__ out, int B, int K) {
  int idx = blockIdx.x * blockDim.x + threadIdx.x;
  if (idx >= B * K) return;
  int b = idx / K;
  int c = cand[idx];
  const float* q = qraw + (long)b * 64;
  const float* g = G + (long)c * 64;
  float s = 0.0f;
#pragma unroll 8
  for (int d = 0; d < 64; ++d) s += (q[d] + b2[d]) * g[d];
  out[idx] = s;
}

// ---------------------------------------------------------------------------
// Launcher
// ---------------------------------------------------------------------------
extern "C" void kernel_launch(void* const* d_in, const int* in_sizes, int n_in,
                              void* d_out, int out_size, void* d_ws, size_t ws_size,
                              hipStream_t stream) {
  const float* query = (const float*)d_in[0];   // (B,768)
  const float* names = (const float*)d_in[1];   // (N,768)
  const float* w1    = (const float*)d_in[2];   // (768,128)
  const float* b1    = (const float*)d_in[3];   // (128)
  const float* w2    = (const float*)d_in[4];   // (128,64)
  const float* b2    = (const float*)d_in[5];   // (64)
  const float* sn_w1 = (const float*)d_in[6];   // (768,256)
  const float* sn_b1 = (const float*)d_in[7];   // (256)
  const float* gam   = (const float*)d_in[8];
  const float* bet   = (const float*)d_in[9];
  const float* mu    = (const float*)d_in[10];
  const float* var   = (const float*)d_in[11];
  const float* sn_w2 = (const float*)d_in[12];  // (256,64)
  const float* sn_b2 = (const float*)d_in[13];  // (64)
  const int*   eidx  = (const int*)d_in[14];    // (2,E)
  const int*   cand  = (const int*)d_in[15];    // (B,Kc)

  const int H    = in_sizes[3];              // 128
  const int D    = in_sizes[5];              // 64
  const int F_IN = in_sizes[2] / H;          // 768
  const int N    = in_sizes[1] / F_IN;       // 50000
  const int B    = in_sizes[0] / F_IN;       // 256
  const int E    = in_sizes[14] / 2;         // 800000
  const int K    = out_size / B;             // 20
  (void)n_in;

  const int* src = eidx;
  const int* dst = eidx + E;

  // Workspace layout (floats)
  float* ws   = (float*)d_ws;
  float* deg  = ws;                 ws += N;
  float* dis  = ws;                 ws += N;
  float* XW1  = ws;                 ws += (size_t)N * H;
  float* AGG1 = ws;                 ws += (size_t)N * H;   // becomes hidden H
  float* XW2  = ws;                 ws += (size_t)N * D;
  float* AGG2 = ws;                 ws += (size_t)N * D;   // becomes names_graph
  float* Z    = ws;                 ws += (size_t)B * 256;
  float* Qr   = ws;                 ws += (size_t)B * D;
  (void)ws_size;

  const int T = 256;
  // 1) degrees + normalization factors
  k_init_deg<<<(N + T - 1) / T, T, 0, stream>>>(deg, N);
  k_count_deg<<<(E + T - 1) / T, T, 0, stream>>>(dst, E, deg);
  k_make_dis<<<(N + T - 1) / T, T, 0, stream>>>(deg, dis, N);

  // 2) conv1: XW1 = names @ w1   (50000x768x128, fp32 WMMA)
  gemm_f32_wmma<768, 128><<<(N + 63) / 64, 256, 0, stream>>>(names, w1, XW1, N);
  (void)hipMemsetAsync(AGG1, 0, (size_t)N * H * sizeof(float), stream);
  {
    long long tot = (long long)E * H;
    k_scatter<128><<<(unsigned)((tot + T - 1) / T), T, 0, stream>>>(src, dst, E, dis, XW1, AGG1);
  }
  k_finish_conv<128, true><<<(unsigned)(((long long)N * H + T - 1) / T), T, 0, stream>>>(
      AGG1, XW1, deg, b1, AGG1, N);

  // 3) conv2: XW2 = Hhid @ w2   (50000x128x64)
  gemm_f32_wmma<128, 64><<<(N + 63) / 64, 128, 0, stream>>>(AGG1, w2, XW2, N);
  (void)hipMemsetAsync(AGG2, 0, (size_t)N * D * sizeof(float), stream);
  {
    long long tot = (long long)E * D;
    k_scatter<64><<<(unsigned)((tot + T - 1) / T), T, 0, stream>>>(src, dst, E, dis, XW2, AGG2);
  }
  k_finish_conv<64, false><<<(unsigned)(((long long)N * D + T - 1) / T), T, 0, stream>>>(
      AGG2, XW2, deg, b2, AGG2, N);

  // 4) query path: Z = query @ sn_w1 ; BN+sigmoid ; Qr = Z @ sn_w2
  gemm_f32_wmma<768, 256><<<(B + 63) / 64, 512, 0, stream>>>(query, sn_w1, Z, B);
  k_bn_sigmoid<<<(B * 256 + T - 1) / T, T, 0, stream>>>(Z, sn_b1, gam, bet, mu, var, B * 256);
  gemm_f32_wmma<256, 64><<<(B + 63) / 64, 128, 0, stream>>>(Z, sn_w2, Qr, B);

  // 5) gather candidates + score (sn_b2 folded into the dot)
  k_score<<<(B * K + T - 1) / T, T, 0, stream>>>(Qr, sn_b2, AGG2, cand, (float*)d_out, B, K);
}